// SO3_ConvNet_VecPrediction_WithInvariantConditioning_18305150615957
// MI455X (gfx1250) — compile-verified
//
#include <hip/hip_runtime.h>

// ============================================================================
// Fused SO3 conv-net forward for MI455X (gfx1250, wave32, WMMA).
//
// Tiling:  grid = B/16 workgroups, 256 threads (8 waves).  A workgroup owns a
// 16-row batch tile; wave w owns output channels [16w, 16w+16).  All channel
// mixing GEMMs are V_WMMA_F32_16X16X4_F32 (full fp32 — matches reference
// precision) with C[o,b] = sum_c W[o,c] * h[c,b]:
//   A (16xK f32, per docs):  lane<16 -> (M=lane, K, K+1), lane>=16 -> (M=lane-16, K+2, K+3)
//   B (Kx16):                lane<16 -> (K,K+1, N=lane),  lane>=16 -> (K+2,K+3, N=lane-16)
//   C (16x16):               lane = column b, VGPR r = row o (r / 8+r per lane half)
// Multi-m GEMM: each weight (A) chunk is loaded once per K-step and reused for
// all 2l+1 m-slices of that degree -> weight VMEM cut by up to 7x vs v1.
// The CG nonlinearity is elementwise in (b, channel) == elementwise in the
// C-tile register layout -> pure VALU, zero cross-lane traffic.  w3j coeffs
// are uniform-address loads -> scalar (s_load) operands to v_fmac.
// LDS: 7 slots x [16 b][128 c] f32 = 56 KB (degrees staged sequentially).
// ============================================================================

typedef float v2f __attribute__((ext_vector_type(2)));
typedef float v4f __attribute__((ext_vector_type(4)));
typedef float v8f __attribute__((ext_vector_type(8)));

#define DEVINL __device__ __forceinline__

constexpr int CH    = 128;
constexpr int SLOT  = 16 * CH;   // one (l,m) slice: 16 batch x 128 channels
constexpr int NSLOT = 7;

// w3j flat-offset table, combos in fixed order (must match generator below)
enum : int {
  OFF_000 = 0,    OFF_011 = 1,    OFF_022 = 10,   OFF_033 = 35,
  OFF_110 = 84,   OFF_112 = 93,   OFF_121 = 138,  OFF_123 = 183,
  OFF_132 = 288,  OFF_220 = 393,  OFF_222 = 418,  OFF_231 = 543,
  OFF_233 = 648,  OFF_330 = 893,  OFF_332 = 942,  W3J_TOTAL = 1187
};

DEVINL v8f vzero() {
  v8f z;
#pragma unroll
  for (int i = 0; i < 8; ++i) z[i] = 0.f;
  return z;
}

// ---- NM 16x16 output tiles sharing one A (weight) operand -----------------
// accs[m] += W(16xCIN) * B_m(CINx16), B_m read from LDS slot m.
template <int CIN, int NM>
DEVINL void gemm_multi(const float* __restrict__ W, const float* bsl0,
                       int wave, int lane, v8f* accs) {
  const int lo = lane & 15, half = lane >> 4;
  const float* ap = W + (wave * 16 + lo) * CIN + 2 * half;  // weights (global)
  const float* bp = bsl0 + lo * CH + 2 * half;              // activations (LDS [b][c])
#pragma unroll
  for (int k = 0; k < CIN; k += 4) {
    v2f a = *reinterpret_cast<const v2f*>(ap + k);          // load A once ...
#pragma unroll
    for (int m = 0; m < NM; ++m) {                          // ... reuse for all m
      v2f b = *reinterpret_cast<const v2f*>(bp + m * SLOT + k);
      accs[m] = __builtin_amdgcn_wmma_f32_16x16x4_f32(false, a, false, b,
                                                      (short)0, accs[m],
                                                      false, false);
    }
  }
}

// ---- scatter a C tile into an LDS slot as [b][c] --------------------------
DEVINL void store_tile(float* slot, v8f t, int wave, int lane, int chbase = 0) {
  const int lo = lane & 15, half = lane >> 4;
  float* dst = slot + lo * CH + chbase + wave * 16 + half * 8;
  v4f a = {t[0], t[1], t[2], t[3]};
  v4f b = {t[4], t[5], t[6], t[7]};
  *reinterpret_cast<v4f*>(dst)     = a;
  *reinterpret_cast<v4f*>(dst + 4) = b;
}

// ---- one CG term: out[l3][k] += w3j[m,n,k] * hl[l1][m] * hl[l2][n] --------
template <int L1, int L2, int L3, int OFF>
DEVINL void cg_term(const float* __restrict__ w3j, const v8f* hl, v8f* acc) {
  constexpr int M1 = 2 * L1 + 1, M2 = 2 * L2 + 1, M3 = 2 * L3 + 1;
  constexpr int S1 = L1 * L1, S2 = L2 * L2, S3 = L3 * L3;
#pragma unroll
  for (int m = 0; m < M1; ++m)
#pragma unroll
    for (int n = 0; n < M2; ++n) {
      v8f p = hl[S1 + m] * hl[S2 + n];
#pragma unroll
      for (int k = 0; k < M3; ++k) {
        float w = w3j[OFF + (m * M2 + n) * M3 + k];  // uniform -> s_load
        acc[S3 + k] += p * w;
      }
    }
}

// ---- one residual block: SO3 linear + CG nonlinearity + skip --------------
template <int LIN, int LOUT>
DEVINL void run_block(v8f* h, const float* const* Wl, const float* __restrict__ w3j,
                      float* smem, int wave, int lane) {
  v8f hl[(LIN + 1) * (LIN + 1)];
#pragma unroll
  for (int l = 0; l <= LIN; ++l) {
    constexpr int LL = 0;  // (placeholder to keep scope clean)
    (void)LL;
#pragma unroll
    for (int m = 0; m < 2 * l + 1; ++m)
      store_tile(smem + m * SLOT, h[l * l + m], wave, lane);
    __syncthreads();
#pragma unroll
    for (int m = 0; m < 2 * l + 1; ++m) hl[l * l + m] = vzero();
    // dispatch on (unrolled-constant) l so NM is a template constant
    if (l == 0)      gemm_multi<CH, 1>(Wl[0], smem, wave, lane, &hl[0]);
    else if (l == 1) gemm_multi<CH, 3>(Wl[1], smem, wave, lane, &hl[1]);
    else if (l == 2) gemm_multi<CH, 5>(Wl[2], smem, wave, lane, &hl[4]);
    else             gemm_multi<CH, 7>(Wl[3], smem, wave, lane, &hl[9]);
    __syncthreads();
  }
  // skip preload (ht[l] += h[l] for l <= LOUT), then CG accumulation
  v8f acc[(LOUT + 1) * (LOUT + 1)];
#pragma unroll
  for (int l = 0; l <= LOUT; ++l)
#pragma unroll
    for (int m = 0; m < 2 * l + 1; ++m) acc[l * l + m] = h[l * l + m];

  cg_term<0, 0, 0, OFF_000>(w3j, hl, acc);
  if constexpr (LIN >= 1 && LOUT >= 1) cg_term<0, 1, 1, OFF_011>(w3j, hl, acc);
  if constexpr (LIN >= 2 && LOUT >= 2) cg_term<0, 2, 2, OFF_022>(w3j, hl, acc);
  if constexpr (LIN >= 3 && LOUT >= 3) cg_term<0, 3, 3, OFF_033>(w3j, hl, acc);
  if constexpr (LIN >= 1)              cg_term<1, 1, 0, OFF_110>(w3j, hl, acc);
  if constexpr (LIN >= 1 && LOUT >= 2) cg_term<1, 1, 2, OFF_112>(w3j, hl, acc);
  if constexpr (LIN >= 2 && LOUT >= 1) cg_term<1, 2, 1, OFF_121>(w3j, hl, acc);
  if constexpr (LIN >= 2 && LOUT >= 3) cg_term<1, 2, 3, OFF_123>(w3j, hl, acc);
  if constexpr (LIN >= 3 && LOUT >= 2) cg_term<1, 3, 2, OFF_132>(w3j, hl, acc);
  if constexpr (LIN >= 2)              cg_term<2, 2, 0, OFF_220>(w3j, hl, acc);
  if constexpr (LIN >= 2 && LOUT >= 2) cg_term<2, 2, 2, OFF_222>(w3j, hl, acc);
  if constexpr (LIN >= 3 && LOUT >= 1) cg_term<2, 3, 1, OFF_231>(w3j, hl, acc);
  if constexpr (LIN >= 3 && LOUT >= 3) cg_term<2, 3, 3, OFF_233>(w3j, hl, acc);
  if constexpr (LIN >= 3)              cg_term<3, 3, 0, OFF_330>(w3j, hl, acc);
  if constexpr (LIN >= 3 && LOUT >= 2) cg_term<3, 3, 2, OFF_332>(w3j, hl, acc);

#pragma unroll
  for (int l = 0; l <= LOUT; ++l)
#pragma unroll
    for (int m = 0; m < 2 * l + 1; ++m) h[l * l + m] = acc[l * l + m];
}

// ============================================================================
// Main fused kernel
// ============================================================================
__global__ __launch_bounds__(256, 1) void so3_fused(
    const float* __restrict__ x0, const float* __restrict__ x1,
    const float* __restrict__ x2, const float* __restrict__ x3,
    const float* __restrict__ cin, const float* __restrict__ w_cond,
    const float* __restrict__ b_cond,
    const float* __restrict__ wi0, const float* __restrict__ wi1,
    const float* __restrict__ wi2, const float* __restrict__ wi3,
    const float* __restrict__ wb0_0, const float* __restrict__ wb0_1,
    const float* __restrict__ wb0_2, const float* __restrict__ wb0_3,
    const float* __restrict__ wb1_0, const float* __restrict__ wb1_1,
    const float* __restrict__ wb1_2, const float* __restrict__ wb1_3,
    const float* __restrict__ wb2_0, const float* __restrict__ wb2_1,
    const float* __restrict__ wb2_2,
    const float* __restrict__ wb3_0, const float* __restrict__ wb3_1,
    const float* __restrict__ w_vec,
    const float* __restrict__ w3j, float* __restrict__ out) {
  __shared__ float smem[NSLOT * SLOT];
  const int tid  = threadIdx.x;
  const int wave = tid >> 5, lane = tid & 31;
  const int b0   = blockIdx.x * 16;

  // ---- stage c -> slot0 [b][0..127]; x0 -> slot1 [b][0..63] ----
  for (int i = tid; i < 16 * 128; i += 256) {
    int b = i >> 7, ch = i & 127;
    smem[0 * SLOT + b * CH + ch] = cin[(b0 + b) * 128 + ch];
  }
  for (int i = tid; i < 16 * 64; i += 256) {
    int b = i >> 6, ch = i & 63;
    smem[1 * SLOT + b * CH + ch] = x0[(b0 + b) * 64 + ch];
  }
  __syncthreads();

  // ---- conditioning MLP: cond[b,j] = c[b,:]·w_cond[j,:] + b_cond[j] ----
  if (wave < 4) {  // 64 output channels -> waves 0..3 (wave-uniform branch)
    v8f t = vzero();
    gemm_multi<128, 1>(w_cond, smem + 0 * SLOT, wave, lane, &t);
    const int half = lane >> 4;
    const int j0 = wave * 16 + half * 8;
#pragma unroll
    for (int r = 0; r < 8; ++r) t[r] += b_cond[j0 + r];
    store_tile(smem + 1 * SLOT, t, wave, lane, /*chbase=*/64);  // concat at ch 64
  }
  __syncthreads();

  // ---- init SO3 linear ----
  v8f h[16];
  h[0] = vzero();
  gemm_multi<128, 1>(wi0, smem + 1 * SLOT, wave, lane, &h[0]);
  __syncthreads();

  // l = 1
  for (int i = tid; i < 16 * 64 * 3; i += 256) {
    int m = i % 3, t2 = i / 3, ch = t2 & 63, b = t2 >> 6;
    smem[m * SLOT + b * CH + ch] = x1[b0 * 192 + i];
  }
  __syncthreads();
#pragma unroll
  for (int m = 0; m < 3; ++m) h[1 + m] = vzero();
  gemm_multi<64, 3>(wi1, smem, wave, lane, &h[1]);
  __syncthreads();

  // l = 2
  for (int i = tid; i < 16 * 64 * 5; i += 256) {
    int m = i % 5, t2 = i / 5, ch = t2 & 63, b = t2 >> 6;
    smem[m * SLOT + b * CH + ch] = x2[b0 * 320 + i];
  }
  __syncthreads();
#pragma unroll
  for (int m = 0; m < 5; ++m) h[4 + m] = vzero();
  gemm_multi<64, 5>(wi2, smem, wave, lane, &h[4]);
  __syncthreads();

  // l = 3
  for (int i = tid; i < 16 * 64 * 7; i += 256) {
    int m = i % 7, t2 = i / 7, ch = t2 & 63, b = t2 >> 6;
    smem[m * SLOT + b * CH + ch] = x3[b0 * 448 + i];
  }
  __syncthreads();
#pragma unroll
  for (int m = 0; m < 7; ++m) h[9 + m] = vzero();
  gemm_multi<64, 7>(wi3, smem, wave, lane, &h[9]);
  __syncthreads();

  // ---- residual blocks ----
  {
    const float* W[4] = {wb0_0, wb0_1, wb0_2, wb0_3};
    run_block<3, 3>(h, W, w3j, smem, wave, lane);
  }
  {
    const float* W[4] = {wb1_0, wb1_1, wb1_2, wb1_3};
    run_block<3, 3>(h, W, w3j, smem, wave, lane);
  }
  {
    const float* W[4] = {wb2_0, wb2_1, wb2_2, nullptr};
    run_block<2, 2>(h, W, w3j, smem, wave, lane);
  }
  {
    const float* W[4] = {wb3_0, wb3_1, nullptr, nullptr};
    run_block<1, 1>(h, W, w3j, smem, wave, lane);
  }

  // ---- vector head: out[b,o,m] = sum_c w_vec[o,c] * h1[b,c,m] ----
#pragma unroll
  for (int m = 0; m < 3; ++m) store_tile(smem + m * SLOT, h[1 + m], wave, lane);
  __syncthreads();
  if (tid < 192) {  // 16 b * 4 o * 3 m
    int b = tid / 12, r = tid % 12, o = r / 3, m = r % 3;
    float s = 0.f;
    for (int ch = 0; ch < 128; ++ch)
      s += smem[m * SLOT + b * CH + ch] * w_vec[o * 128 + ch];
    out[((b0 + b) * 4 + o) * 3 + m] = s;
  }
}

// ============================================================================
// Device-side exact reproduction of np.random.RandomState(seed).randn(...)
// (MT19937 init_genrand + rk_double + polar-method Gaussian), used to build
// the deterministic stand-in Clebsch-Gordan tensors.  Single thread, ~3k MT
// draws — negligible, graph-capture safe, no host memory involved.
// ============================================================================
struct RS {
  unsigned mt[624];
  int mti;
  bool has_g;
  double g;
};
DEVINL void rs_init(RS& s, unsigned seed) {
  s.mt[0] = seed;
  for (int i = 1; i < 624; ++i)
    s.mt[i] = 1812433253u * (s.mt[i - 1] ^ (s.mt[i - 1] >> 30)) + (unsigned)i;
  s.mti = 624;
  s.has_g = false;
  s.g = 0.0;
}
DEVINL unsigned rs_u32(RS& s) {
  if (s.mti >= 624) {
    for (int i = 0; i < 624; ++i) {
      unsigned y = (s.mt[i] & 0x80000000u) | (s.mt[(i + 1) % 624] & 0x7fffffffu);
      s.mt[i] = s.mt[(i + 397) % 624] ^ (y >> 1) ^ ((y & 1u) ? 2567483615u : 0u);
    }
    s.mti = 0;
  }
  unsigned y = s.mt[s.mti++];
  y ^= y >> 11;
  y ^= (y << 7) & 2636928640u;
  y ^= (y << 15) & 4022730752u;
  y ^= y >> 18;
  return y;
}
DEVINL double rs_dbl(RS& s) {
  unsigned a = rs_u32(s) >> 5, b = rs_u32(s) >> 6;
  return (a * 67108864.0 + b) / 9007199254740992.0;
}
DEVINL double rs_gauss(RS& s) {
  if (s.has_g) { s.has_g = false; return s.g; }
  double x1, x2, r2;
  do {
    x1 = 2.0 * rs_dbl(s) - 1.0;
    x2 = 2.0 * rs_dbl(s) - 1.0;
    r2 = x1 * x1 + x2 * x2;
  } while (r2 >= 1.0 || r2 == 0.0);
  double f = sqrt(-2.0 * log(r2) / r2);
  s.g = f * x1;
  s.has_g = true;
  return f * x2;
}

__global__ void gen_w3j(float* __restrict__ w3j) {
  if (threadIdx.x != 0 || blockIdx.x != 0) return;
  const int combos[15][3] = {{0,0,0},{0,1,1},{0,2,2},{0,3,3},{1,1,0},
                             {1,1,2},{1,2,1},{1,2,3},{1,3,2},{2,2,0},
                             {2,2,2},{2,3,1},{2,3,3},{3,3,0},{3,3,2}};
  RS s;
  int off = 0;
  for (int ci = 0; ci < 15; ++ci) {
    const int l1 = combos[ci][0], l2 = combos[ci][1], l3 = combos[ci][2];
    const int n = (2 * l1 + 1) * (2 * l2 + 1) * (2 * l3 + 1);
    rs_init(s, (unsigned)(100 * l1 + 10 * l2 + l3));
    const double norm = sqrt((double)((2 * l1 + 1) * (2 * l2 + 1)));
    for (int i = 0; i < n; ++i) {
      float g = (float)rs_gauss(s);              // randn(...).astype(float32)
      w3j[off + i] = (float)((double)g / norm);  // then / sqrt(...)
    }
    off += n;
  }
}

// ============================================================================
extern "C" void kernel_launch(void* const* d_in, const int* in_sizes, int n_in,
                              void* d_out, int out_size, void* d_ws, size_t ws_size,
                              hipStream_t stream) {
  const float* x0     = (const float*)d_in[0];
  const float* x1     = (const float*)d_in[1];
  const float* x2     = (const float*)d_in[2];
  const float* x3     = (const float*)d_in[3];
  const float* cin    = (const float*)d_in[4];
  const float* w_cond = (const float*)d_in[5];
  const float* b_cond = (const float*)d_in[6];
  const float* wi0    = (const float*)d_in[7];
  const float* wi1    = (const float*)d_in[8];
  const float* wi2    = (const float*)d_in[9];
  const float* wi3    = (const float*)d_in[10];
  const float* wb0_0  = (const float*)d_in[11];
  const float* wb0_1  = (const float*)d_in[12];
  const float* wb0_2  = (const float*)d_in[13];
  const float* wb0_3  = (const float*)d_in[14];
  const float* wb1_0  = (const float*)d_in[15];
  const float* wb1_1  = (const float*)d_in[16];
  const float* wb1_2  = (const float*)d_in[17];
  const float* wb1_3  = (const float*)d_in[18];
  const float* wb2_0  = (const float*)d_in[19];
  const float* wb2_1  = (const float*)d_in[20];
  const float* wb2_2  = (const float*)d_in[21];
  const float* wb3_0  = (const float*)d_in[22];
  const float* wb3_1  = (const float*)d_in[23];
  const float* w_vec  = (const float*)d_in[24];

  float* w3j = (float*)d_ws;  // 1187 floats of scratch

  const int B = in_sizes[4] / 128;  // c is [B, 128]
  gen_w3j<<<1, 1, 0, stream>>>(w3j);
  so3_fused<<<B / 16, 256, 0, stream>>>(
      x0, x1, x2, x3, cin, w_cond, b_cond, wi0, wi1, wi2, wi3,
      wb0_0, wb0_1, wb0_2, wb0_3, wb1_0, wb1_1, wb1_2, wb1_3,
      wb2_0, wb2_1, wb2_2, wb3_0, wb3_1, w_vec, w3j, (float*)d_out);
}